// Equalize_74577812128469
// MI455X (gfx1250) — compile-verified
//
#include <hip/hip_runtime.h>
#include <hip/hip_bf16.h>

// Problem geometry (fixed by the reference): x is (64, 3, 512, 512) fp32.
#define BC   192          // B*C channels
#define HW   262144       // 512*512 pixels per channel
#define HW4  65536        // HW / 4 (float4 granules per channel)
#define BPC  16           // blocks per channel for streaming passes
#define TPB  256          // threads per block (8 wave32 waves)
#define ITERS (HW4 / (BPC * TPB))   // 16 float4 loads per thread

typedef __attribute__((ext_vector_type(2))) float v2f;
typedef __attribute__((ext_vector_type(8))) float v8f;

// ---------------------------------------------------------------- pass 0: zero
__global__ void eq_zero_hist(int* __restrict__ hist) {
    int i = blockIdx.x * blockDim.x + threadIdx.x;
    if (i < BC * 256) hist[i] = 0;
}

// ------------------------------------------------------------ pass 1: histogram
// One block = one 16K-pixel slice of one channel. 8 per-wave LDS sub-histograms
// (wave32 -> only 32 lanes contend per copy; ds_add_u32 atomics), merged into the
// per-channel global histogram with one global atomic per bin per block.
__global__ __launch_bounds__(TPB) void eq_hist(const float* __restrict__ x,
                                               int* __restrict__ hist) {
    __shared__ int sub[8 * 256];
    const int tid  = threadIdx.x;
    const int wave = tid >> 5;
    #pragma unroll
    for (int i = tid; i < 8 * 256; i += TPB) sub[i] = 0;
    __syncthreads();

    const int c     = blockIdx.x / BPC;
    const int chunk = blockIdx.x % BPC;
    const float4* p = reinterpret_cast<const float4*>(x)
                    + (size_t)c * HW4 + (size_t)chunk * (HW4 / BPC);
    int* h = sub + wave * 256;

    for (int i = 0; i < ITERS; ++i) {
        float4 v = p[i * TPB + tid];                        // global_load_b128
        int b0 = min(max((int)(v.x * 255.0f), 0), 255);
        int b1 = min(max((int)(v.y * 255.0f), 0), 255);
        int b2 = min(max((int)(v.z * 255.0f), 0), 255);
        int b3 = min(max((int)(v.w * 255.0f), 0), 255);
        atomicAdd(&h[b0], 1);                               // ds_add_u32
        atomicAdd(&h[b1], 1);
        atomicAdd(&h[b2], 1);
        atomicAdd(&h[b3], 1);
    }
    __syncthreads();

    // tid indexes one bin (TPB == 256): merge 8 sub-histograms, one global atomic.
    int s = 0;
    #pragma unroll
    for (int w = 0; w < 8; ++w) s += sub[w * 256 + tid];
    if (s) atomicAdd(&hist[c * 256 + tid], s);
}

// ------------------------------------------------- pass 2: LUT via WMMA cumsum
// One wave32 per channel. hist viewed as 16x16 matrix H (element (m,n) =
// hist[16m+n]). Row-wise inclusive prefix = H x U (U = upper-tri ones), computed
// as 4 chained V_WMMA_F32_16X16X4_F32 (K = 16 split into 4x4). Cross-row carry
// (16 row totals) finished with a tiny LDS scan. Counts <= 2^18 are exact in f32.
__global__ __launch_bounds__(32) void eq_lut(const int* __restrict__ hist,
                                             float* __restrict__ lutf) {
    __shared__ int   histS[256];
    __shared__ int   cumS[256];
    __shared__ float rowsumS[16];
    __shared__ int   stepS;

    const int lane = threadIdx.x;          // full wave, EXEC all ones for WMMA
    const int c    = blockIdx.x;
    const int* h   = hist + c * 256;
    #pragma unroll
    for (int t = 0; t < 8; ++t) histS[lane + 32 * t] = h[lane + 32 * t];
    __syncthreads();

    const int hi = lane >> 4;              // lane half selects K-subgroup
    const int m  = lane & 15;              // A: row index; B: column index

    v8f acc = {};
    #pragma unroll
    for (int kc = 0; kc < 4; ++kc) {
        const int k0 = 4 * kc + 2 * hi;    // ISA f32 A/B fragment K layout
        v2f a, b;
        a.x = (float)histS[16 * m + k0 + 0];      // A = H (16x4 chunk)
        a.y = (float)histS[16 * m + k0 + 1];
        b.x = (k0 + 0 <= m) ? 1.0f : 0.0f;        // B = U rows k0..k0+1 (4x16 chunk)
        b.y = (k0 + 1 <= m) ? 1.0f : 0.0f;
        acc = __builtin_amdgcn_wmma_f32_16x16x4_f32(
            /*neg_a=*/false, a, /*neg_b=*/false, b,
            /*c_mod=*/(short)0, acc, /*reuse_a=*/false, /*reuse_b=*/false);
    }
    // acc[r] = prefix of row M = r + 8*hi up to column N = lane&15.

    if ((lane & 15) == 15) {               // column 15 holds full row sums
        #pragma unroll
        for (int r = 0; r < 8; ++r) rowsumS[8 * hi + r] = acc[r];
    }
    __syncthreads();

    float roff[8];
    float run = 0.0f;
    #pragma unroll
    for (int k = 0; k < 16; ++k) {         // exclusive prefix over row totals
        if ((k >> 3) == hi) roff[k & 7] = run;
        run += rowsumS[k];
    }
    const int n = lane & 15;
    #pragma unroll
    for (int r = 0; r < 8; ++r) {
        int M = 8 * hi + r;
        cumS[16 * M + n] = (int)(acc[r] + roff[r] + 0.5f);  // exact integers
    }
    __syncthreads();

    if (lane == 0) {                       // last nonzero bin -> step
        int idx = 255;
        while (idx > 0 && histS[idx] == 0) --idx;
        stepS = (HW - histS[idx]) / 255;
    }
    __syncthreads();

    const int step = stepS;
    float* lo = lutf + c * 256;
    #pragma unroll
    for (int t = 0; t < 8; ++t) {
        int i = lane + 32 * t;
        int v;
        if (step == 0)      v = i;                         // identity channel
        else if (i == 0)    v = 0;                         // shifted-by-one LUT
        else {
            v = (cumS[i - 1] + (step >> 1)) / step;
            v = v < 0 ? 0 : (v > 255 ? 255 : v);
        }
        lo[i] = (float)v * (1.0f / 255.0f);                // pre-divide by 255
    }
}

// ------------------------------------------------------------- pass 3: apply
// Streaming gather: channel LUT staged in LDS, float4 in, float4 out.
__global__ __launch_bounds__(TPB) void eq_apply(const float* __restrict__ x,
                                                const float* __restrict__ lutf,
                                                float* __restrict__ out) {
    __shared__ float l[256];
    const int tid   = threadIdx.x;
    const int c     = blockIdx.x / BPC;
    const int chunk = blockIdx.x % BPC;
    l[tid] = lutf[c * 256 + tid];
    __syncthreads();

    const size_t base = (size_t)c * HW4 + (size_t)chunk * (HW4 / BPC);
    const float4* p = reinterpret_cast<const float4*>(x) + base;
    float4*       q = reinterpret_cast<float4*>(out) + base;

    for (int i = 0; i < ITERS; ++i) {
        float4 v = p[i * TPB + tid];
        float4 r;
        r.x = l[min(max((int)(v.x * 255.0f), 0), 255)];
        r.y = l[min(max((int)(v.y * 255.0f), 0), 255)];
        r.z = l[min(max((int)(v.z * 255.0f), 0), 255)];
        r.w = l[min(max((int)(v.w * 255.0f), 0), 255)];
        q[i * TPB + tid] = r;
    }
}

// ---------------------------------------------------------------------- launch
extern "C" void kernel_launch(void* const* d_in, const int* in_sizes, int n_in,
                              void* d_out, int out_size, void* d_ws, size_t ws_size,
                              hipStream_t stream) {
    const float* x   = (const float*)d_in[0];
    float*       out = (float*)d_out;
    int*   hist = (int*)d_ws;                                   // 192*256 int32
    float* lutf = (float*)((char*)d_ws + BC * 256 * sizeof(int)); // 192*256 f32

    eq_zero_hist<<<BC, 256, 0, stream>>>(hist);
    eq_hist <<<BC * BPC, TPB, 0, stream>>>(x, hist);
    eq_lut  <<<BC, 32, 0, stream>>>(hist, lutf);
    eq_apply<<<BC * BPC, TPB, 0, stream>>>(x, lutf, out);
}